// TopKRouter_70334384439374
// MI455X (gfx1250) — compile-verified
//
#include <hip/hip_runtime.h>
#include <hip/hip_bf16.h>
#include <math.h>

// MoE top-2 router for MI455X (gfx1250, wave32).
// Fused: logits GEMM (V_WMMA_F32_16X16X4_F32) + fp32 softmax + top-2 +
// aux/z loss partials. Memory-bound (~257MB @ 23.3TB/s ~ 11us):
//  - hs streamed exactly once via GLOBAL_LOAD_ASYNC_TO_LDS_B128 (ASYNCcnt),
//    double-buffered so chunk c+1's DMA overlaps chunk c's WMMAs
//  - gate_w (1MB) stays L2-resident
//  - logits never touch HBM (LDS-fused softmax/top-2/loss partials)

typedef __attribute__((ext_vector_type(2))) float v2f;
typedef __attribute__((ext_vector_type(8))) float v8f;

#define NUM_EXPERTS 64
#define HDIM 4096
#define KC 64             // K-chunk staged in LDS
#define NCHUNK (HDIM / KC)
#define TOK_PER_BLOCK 128 // 8 waves x 16 tokens
#define LSTR 68           // padded LDS row stride (floats) for hs/gate tiles
#define LOGSTR 65         // padded LDS row stride (floats) for logits
#define BUFSZ (TOK_PER_BLOCK * LSTR + NUM_EXPERTS * LSTR)  // floats per buffer

__device__ __forceinline__ unsigned int lds_byte_off(const void* p) {
    // generic -> local addrspacecast strips the LDS aperture, leaving the
    // raw byte offset that GLOBAL_LOAD_ASYNC_TO_LDS's VDST operand expects
    return (unsigned int)(unsigned long long)
        (const __attribute__((address_space(3))) void*)p;
}

__device__ __forceinline__ void async_ld_b128(unsigned int ldsOff, const float* g) {
    asm volatile("global_load_async_to_lds_b128 %0, %1, off"
                 :: "v"(ldsOff), "v"((unsigned long long)(uintptr_t)g)
                 : "memory");
}

__global__ __launch_bounds__(256) void router_fused_kernel(
    const float* __restrict__ hs,    // [T, H]
    const float* __restrict__ gate,  // [E, H]
    float* __restrict__ rw_out,      // [T, 2]
    int*   __restrict__ sel_out,     // [T, 2]
    float* __restrict__ ws_prob,     // [64] sum of probs per expert
    float* __restrict__ ws_cnt,      // [64] top-2 counts per expert
    float* __restrict__ ws_z)        // [1]  sum lse^2
{
    __shared__ __align__(16) float smem[2 * BUFSZ];
    __shared__ float expProb[NUM_EXPERTS];
    __shared__ float expCnt[NUM_EXPERTS];
    __shared__ float zAcc;

    const int tid  = threadIdx.x;
    const int lane = tid & 31;
    const int wave = tid >> 5;
    const int tokenBase = blockIdx.x * TOK_PER_BLOCK;

    if (tid < NUM_EXPERTS) { expProb[tid] = 0.0f; expCnt[tid] = 0.0f; }
    if (tid == 0) zAcc = 0.0f;

    // ---- staging geometry (each thread owns 8 hs + 4 gate b128 transfers) ----
    const int hsRow[2] = { tid >> 4, 0 };       // placeholder to keep arrays off
    const int rowT = tid >> 4;                  // used below via idx math
    (void)hsRow; (void)rowT;

    // per-buffer LDS byte bases
    unsigned int hsBase[2], gwBase[2];
    hsBase[0] = lds_byte_off(&smem[0]);
    gwBase[0] = lds_byte_off(&smem[TOK_PER_BLOCK * LSTR]);
    hsBase[1] = lds_byte_off(&smem[BUFSZ]);
    gwBase[1] = lds_byte_off(&smem[BUFSZ + TOK_PER_BLOCK * LSTR]);

    // issue all async loads for chunk (kBase) into buffer `b`
    auto issue_chunk = [&](int kBase, int b) {
        #pragma unroll
        for (int i = 0; i < 8; ++i) {               // hs: 2048 b128 / 256 thr
            int idx = tid + i * 256;
            int row = idx >> 4, c4 = idx & 15;
            async_ld_b128(hsBase[b] + (unsigned)(row * LSTR + c4 * 4) * 4u,
                          hs + (size_t)(tokenBase + row) * HDIM + kBase + c4 * 4);
        }
        #pragma unroll
        for (int i = 0; i < 4; ++i) {               // gate: 1024 b128 / 256 thr
            int idx = tid + i * 256;
            int row = idx >> 4, c4 = idx & 15;
            async_ld_b128(gwBase[b] + (unsigned)(row * LSTR + c4 * 4) * 4u,
                          gate + (size_t)row * HDIM + kBase + c4 * 4);
        }
    };

    // ---- WMMA fragment geometry (f32 16x16x4) ----
    const int m    = lane & 15;
    const int kofs = (lane >> 4) * 2;
    const int waveRow = wave * 16;

    v8f acc0 = {}, acc1 = {}, acc2 = {}, acc3 = {};

    issue_chunk(0, 0);                              // prologue: chunk 0 -> buf 0

    for (int c = 0; c < NCHUNK; ++c) {
        const int cur = c & 1;
        if (c + 1 < NCHUNK) {
            issue_chunk((c + 1) * KC, cur ^ 1);     // overlap next chunk's DMA
            asm volatile("s_wait_asynccnt 0xc" ::: "memory");  // chunk c done
        } else {
            asm volatile("s_wait_asynccnt 0x0" ::: "memory");
        }
        __syncthreads();                            // chunk c visible to all waves

        float* hsT = smem + cur * BUFSZ;
        float* gw  = hsT + TOK_PER_BLOCK * LSTR;
        const float* aRow = hsT + (waveRow + m) * LSTR + kofs;
        const float* b0 = gw + ( 0 + m) * LSTR + kofs;
        const float* b1 = gw + (16 + m) * LSTR + kofs;
        const float* b2 = gw + (32 + m) * LSTR + kofs;
        const float* b3 = gw + (48 + m) * LSTR + kofs;

        // software-pipelined: prefetch k+4 fragments before k's WMMAs
        v2f a  = *(const v2f*)(aRow);
        v2f f0 = *(const v2f*)(b0);
        v2f f1 = *(const v2f*)(b1);
        v2f f2 = *(const v2f*)(b2);
        v2f f3 = *(const v2f*)(b3);
        #pragma unroll
        for (int kk = 0; kk < KC; kk += 4) {
            v2f an, g0, g1, g2, g3;
            if (kk + 4 < KC) {
                an = *(const v2f*)(aRow + kk + 4);
                g0 = *(const v2f*)(b0 + kk + 4);
                g1 = *(const v2f*)(b1 + kk + 4);
                g2 = *(const v2f*)(b2 + kk + 4);
                g3 = *(const v2f*)(b3 + kk + 4);
            }
            acc0 = __builtin_amdgcn_wmma_f32_16x16x4_f32(
                false, a, false, f0, (short)0, acc0, false, false);
            acc1 = __builtin_amdgcn_wmma_f32_16x16x4_f32(
                false, a, false, f1, (short)0, acc1, false, false);
            acc2 = __builtin_amdgcn_wmma_f32_16x16x4_f32(
                false, a, false, f2, (short)0, acc2, false, false);
            acc3 = __builtin_amdgcn_wmma_f32_16x16x4_f32(
                false, a, false, f3, (short)0, acc3, false, false);
            if (kk + 4 < KC) { a = an; f0 = g0; f1 = g1; f2 = g2; f3 = g3; }
        }
        __syncthreads();   // protect buf[cur^0] before it's refilled at c+2
    }

    // Spill logits to LDS (alias buffer 0's hs region)
    float* lgt = smem;
    {
        const int half = (lane >> 4) * 8;
        #pragma unroll
        for (int r = 0; r < 8; ++r) {
            int tr = waveRow + r + half;
            lgt[tr * LOGSTR +  0 + m] = acc0[r];
            lgt[tr * LOGSTR + 16 + m] = acc1[r];
            lgt[tr * LOGSTR + 32 + m] = acc2[r];
            lgt[tr * LOGSTR + 48 + m] = acc3[r];
        }
    }
    __syncthreads();

    // Per-token softmax + top-2 + loss partials (one thread per token)
    if (tid < TOK_PER_BLOCK) {
        const float* L = lgt + tid * LOGSTR;

        float mx = L[0];
        #pragma unroll
        for (int e = 1; e < NUM_EXPERTS; ++e) mx = fmaxf(mx, L[e]);

        float sum = 0.0f;
        float l1 = -INFINITY, l2 = -INFINITY;
        int   i1 = 0, i2 = 0;
        for (int e = 0; e < NUM_EXPERTS; ++e) {
            float l = L[e];
            sum += __expf(l - mx);
            if (l > l1)      { l2 = l1; i2 = i1; l1 = l; i1 = e; }
            else if (l > l2) { l2 = l;  i2 = e; }
        }
        float invSum = 1.0f / sum;
        float e1 = __expf(l1 - mx);
        float e2 = __expf(l2 - mx);
        float wden = 1.0f / (e1 + e2);
        float lse = mx + __logf(sum);

        for (int i = 0; i < NUM_EXPERTS; ++i) {      // rotated to dodge hotspots
            int e = (tid + i) & (NUM_EXPERTS - 1);
            atomicAdd(&expProb[e], __expf(L[e] - mx) * invSum);
        }
        atomicAdd(&expCnt[i1], 1.0f);
        atomicAdd(&expCnt[i2], 1.0f);
        atomicAdd(&zAcc, lse * lse);

        int g = tokenBase + tid;
        rw_out[2 * g + 0] = e1 * wden;
        rw_out[2 * g + 1] = e2 * wden;
        sel_out[2 * g + 0] = i1;
        sel_out[2 * g + 1] = i2;
    }

    __syncthreads();
    if (tid < NUM_EXPERTS) {
        atomicAdd(&ws_prob[tid], expProb[tid]);
        atomicAdd(&ws_cnt[tid],  expCnt[tid]);
    }
    if (tid == 0) atomicAdd(ws_z, zAcc);
}

__global__ void router_loss_kernel(const float* __restrict__ ws_prob,
                                   const float* __restrict__ ws_cnt,
                                   const float* __restrict__ ws_z,
                                   float* __restrict__ loss_out,
                                   float numTokens)
{
    if (threadIdx.x == 0 && blockIdx.x == 0) {
        float aux = 0.0f;
        for (int e = 0; e < NUM_EXPERTS; ++e)
            aux += (ws_cnt[e] / numTokens) * (ws_prob[e] / numTokens);
        aux *= (float)NUM_EXPERTS;
        float z = ws_z[0] / numTokens;
        loss_out[0] = 0.01f * aux + 0.001f * z;
    }
}

extern "C" void kernel_launch(void* const* d_in, const int* in_sizes, int n_in,
                              void* d_out, int out_size, void* d_ws, size_t ws_size,
                              hipStream_t stream) {
    const float* hs   = (const float*)d_in[0];  // [4,4096,4096] fp32
    const float* gate = (const float*)d_in[1];  // [64,4096] fp32

    const int H = HDIM;
    const int T = in_sizes[0] / H;              // 16384 tokens

    float* out  = (float*)d_out;
    float* rw   = out;                          // [T,2] fp32
    int*   sel  = (int*)(out + (size_t)T * 2);  // [T,2] int32
    float* loss = out + (size_t)T * 4;          // [1]

    float* wsp = (float*)d_ws;      // [64] prob sums
    float* wsc = wsp + 64;          // [64] counts
    float* wsz = wsp + 128;         // [1]  z partial

    hipMemsetAsync(d_ws, 0, 129 * sizeof(float), stream);

    dim3 grid(T / TOK_PER_BLOCK);
    router_fused_kernel<<<grid, 256, 0, stream>>>(hs, gate, rw, sel, wsp, wsc, wsz);
    router_loss_kernel<<<1, 64, 0, stream>>>(wsp, wsc, wsz, loss, (float)T);
}